// YOLOLoss_22737556865663
// MI455X (gfx1250) — compile-verified
//
#include <hip/hip_runtime.h>

#define NBANCH 10
#define NCLS   14
#define LCH    19
#define NOBJ   256
#define BDIM   8
#define HDIM   96
#define WDIM   96
#define CDIM   (NBANCH * LCH)      // 190
#define PLANE  (HDIM * WDIM)       // 9216
#define NMASK  (BDIM * NBANCH * PLANE) // 737280

typedef float v2f __attribute__((ext_vector_type(2)));
typedef float v8f __attribute__((ext_vector_type(8)));

__device__ __forceinline__ int pidx(int b, int c, int h, int w) {
    return ((b * CDIM + c) * HDIM + h) * WDIM + w;
}

// ---------------- zero the mask array (word-wise) ----------------
__global__ void k_zero(unsigned int* w, int nwords) {
    int i = blockIdx.x * blockDim.x + threadIdx.x;
    if (i < nwords) w[i] = 0u;
}

// ---------------- IoU between target box and the 10 anchor pred boxes ----------------
__global__ void k_iou(const float* __restrict__ P, const float* __restrict__ T,
                      const int* __restrict__ grids, float* __restrict__ ious) {
    int t = blockIdx.x * blockDim.x + threadIdx.x;
    if (t >= NOBJ * NBANCH) return;
    int n = t / NBANCH, a = t % NBANCH;
    int b = grids[n*4+0], g = grids[n*4+1], h = grids[n*4+2], w = grids[n*4+3];

    float tx = T[pidx(b, g*LCH+1, h, w)];
    float ty = T[pidx(b, g*LCH+2, h, w)];
    float tw = T[pidx(b, g*LCH+3, h, w)];
    float th = T[pidx(b, g*LCH+4, h, w)];
    float px = P[pidx(b, a*LCH+1, h, w)];
    float py = P[pidx(b, a*LCH+2, h, w)];
    float pw = P[pidx(b, a*LCH+3, h, w)];
    float ph = P[pidx(b, a*LCH+4, h, w)];

    float tx1 = tx - tw*0.5f, ty1 = ty - th*0.5f, tx2 = tx + tw*0.5f, ty2 = ty + th*0.5f;
    float px1 = px - pw*0.5f, py1 = py - ph*0.5f, px2 = px + pw*0.5f, py2 = py + ph*0.5f;

    float x1 = fmaxf(tx1, px1), y1 = fmaxf(ty1, py1);
    float x2 = fminf(tx2, px2), y2 = fminf(ty2, py2);
    float inter = fmaxf(x2 - x1, 0.0f) * fmaxf(y2 - y1, 0.0f);
    float a1 = fabsf((tx2 - tx1) * (ty2 - ty1));
    float a2 = fabsf((px2 - px1) * (py2 - py1));
    ious[n * NBANCH + a] = inter / (a1 + a2 - inter + 1e-6f);
}

// ---------------- sequential responsible-anchor scan (1 wave, LDS-resident) ----------------
// maxI is only written at the 256 grid cells; the row read mI[gb,gh,gw,:] sees live
// values only at grids j with (b_j,g_j,h_j)==(gb,gh,gw); uncovered slots stay 0.
__global__ void k_scan(const int* __restrict__ grids, const float* __restrict__ ious,
                       int* __restrict__ sel, int* __restrict__ selxy) {
    __shared__ int Gb[NOBJ], Gg[NOBJ], Gh[NOBJ], Gw[NOBJ];
    __shared__ int cv[NOBJ];
    int t = threadIdx.x; // 0..31
    for (int j = t; j < NOBJ; j += 32) {
        Gb[j] = grids[j*4+0]; Gg[j] = grids[j*4+1];
        Gh[j] = grids[j*4+2]; Gw[j] = grids[j*4+3];
        cv[j] = 0;
    }
    __syncthreads();

    int base = t * 8;
    for (int i = 0; i < NOBJ; ++i) {
        int gb = Gb[i], gh = Gh[i], gw = Gw[i];
        // static match set for this row (values change, geometry doesn't)
        int matchBits = 0;
        for (int k = 0; k < 8; ++k) {
            int j = base + k;
            if (Gb[j] == gb && Gg[j] == gh && Gh[j] == gw) matchBits |= (1 << k);
        }
        int cov = __popc(matchBits);
        for (int off = 16; off > 0; off >>= 1) cov += __shfl_down(cov, off);
        cov = __shfl(cov, 0); // distinct w per match (grids unique) => cov == covered slots

        for (int a = 0; a < NBANCH; ++a) {
            int cur = cv[i];
            float iou = ious[i * NBANCH + a];
            bool hasA = false;
            int mb = matchBits;
            while (mb) {
                int k = __ffs(mb) - 1; mb &= mb - 1;
                if (cv[base + k] == a) { hasA = true; break; }
            }
            bool anyA  = (__ballot(hasA) != 0ull);
            // all(row != a): no live cell equals a, and if a==0 also no uncovered zero slot
            bool allNot = !anyA && ((a != 0) || (cov == WDIM));
            bool cond   = (((float)cur) < iou) && allNot;
            __syncthreads();
            if (t == 0) cv[i] = cond ? a : cur;
            __syncthreads();
        }
    }
    __syncthreads();

    for (int i = t; i < NOBJ; i += 32) {
        sel[i] = cv[i];
        // sel_xy = maxI[b, g, h, h]  (reference quirk: W index = h)
        int b = Gb[i], g = Gg[i], h = Gh[i];
        int v = 0;
        for (int j = 0; j < NOBJ; ++j)
            if (Gb[j] == b && Gg[j] == g && Gh[j] == h && Gw[j] == h) v = cv[j];
        selxy[i] = v;
    }
}

// ---------------- per-grid losses + mask scatter (one 256-thread block) ----------------
__global__ void k_gridloss(const float* __restrict__ P, const float* __restrict__ T,
                           const int* __restrict__ grids,
                           const int* __restrict__ sel, const int* __restrict__ selxy,
                           unsigned char* __restrict__ mask, float* __restrict__ accum) {
    __shared__ float sLoc[NOBJ], sConf[NOBJ], sCls[NOBJ];
    int i = threadIdx.x;
    int b = grids[i*4+0], g = grids[i*4+1], h = grids[i*4+2], w = grids[i*4+3];
    int s = sel[i], sxy = selxy[i];

    mask[((b * NBANCH + s) * HDIM + h) * WDIM + w] = 1;

    float tx = T[pidx(b, g*LCH+1, h, w)];
    float ty = T[pidx(b, g*LCH+2, h, w)];
    float tw = T[pidx(b, g*LCH+3, h, w)];
    float th = T[pidx(b, g*LCH+4, h, w)];
    float px = P[pidx(b, sxy*LCH+1, h, w)];
    float py = P[pidx(b, sxy*LCH+2, h, w)];
    float pw = P[pidx(b, s*LCH+3, h, w)];
    float ph = P[pidx(b, s*LCH+4, h, w)];

    float dx = tx - px, dy = ty - py;
    float dw = sqrtf(fmaxf(tw, 0.0f)) - sqrtf(fmaxf(pw, 0.0f));
    float dh = sqrtf(fmaxf(th, 0.0f)) - sqrtf(fmaxf(ph, 0.0f));
    sLoc[i] = dx*dx + dy*dy + dw*dw + dh*dh;

    float dc = T[pidx(b, g*LCH, h, w)] - P[pidx(b, s*LCH, h, w)];
    sConf[i] = dc * dc;

    float cs = 0.0f;
    for (int c = 0; c < NCLS; ++c) {
        float d = P[pidx(b, s*LCH + 5 + c, h, w)] - T[pidx(b, b*LCH + 5 + c, h, w)];
        cs += d * d;
    }
    sCls[i] = cs * (1.0f / NCLS);
    __syncthreads();

    for (int off = 128; off > 0; off >>= 1) {
        if (i < off) {
            sLoc[i]  += sLoc[i + off];
            sConf[i] += sConf[i + off];
            sCls[i]  += sCls[i + off];
        }
        __syncthreads();
    }
    if (i == 0) { accum[0] = sLoc[0]; accum[1] = sConf[0]; accum[2] = sCls[0]; }
}

// ---------------- masked Σ(tc-pc)^2 over all (b,a) confidence planes ----------------
// One block per plane (80 planes x 9216 f32). Wave reduce via V_WMMA_F32_16X16X4_F32
// with B = ones: D rows broadcast row-sums of A; exact f32 arithmetic (products x1.0).
__global__ void k_bigred(const float* __restrict__ P, const float* __restrict__ T,
                         const unsigned char* __restrict__ mask,
                         float* __restrict__ blockSums) {
    int p = blockIdx.x;                 // 0..79  = b*10 + a
    int b = p / NBANCH, a = p % NBANCH;
    const float4* Tp = (const float4*)(T + (size_t)pidx(b, a * LCH, 0, 0));
    const float4* Pp = (const float4*)(P + (size_t)pidx(b, a * LCH, 0, 0));
    const uchar4* Mp = (const uchar4*)(mask + (size_t)p * PLANE);

    float acc = 0.0f;
    const int N4 = PLANE / 4;           // 2304 = 9 * 256, no divergence
    for (int e = threadIdx.x; e < N4; e += 256) {
        if (e + 512 < N4) {             // speculative prefetch of upcoming lines
            __builtin_prefetch(&Tp[e + 512], 0, 0);
            __builtin_prefetch(&Pp[e + 512], 0, 0);
        }
        float4 tv = Tp[e];
        float4 pv = Pp[e];
        uchar4 mv = Mp[e];
        float d0 = tv.x - pv.x, d1 = tv.y - pv.y, d2 = tv.z - pv.z, d3 = tv.w - pv.w;
        acc += (mv.x ? 0.0f : d0 * d0);
        acc += (mv.y ? 0.0f : d1 * d1);
        acc += (mv.z ? 0.0f : d2 * d2);
        acc += (mv.w ? 0.0f : d3 * d3);
    }

    float waveSum;
#if __has_builtin(__builtin_amdgcn_wmma_f32_16x16x4_f32)
    // A 16x4 f32: lanes 0-15 -> K=0(v0),K=1(v1); lanes 16-31 -> K=2,K=3. Put partial in v0.
    v2f A;  A.x = acc;  A.y = 0.0f;
    v2f Bm; Bm.x = 1.0f; Bm.y = 1.0f;
    v8f C = {};
    v8f D = __builtin_amdgcn_wmma_f32_16x16x4_f32(false, A, false, Bm, (short)0, C, false, false);
    // lane L<16 holds D[v][L] (v=0..7 -> M=0..7); lane L>=16 holds D[v+8][L-16].
    float s = D[0] + D[1] + D[2] + D[3] + D[4] + D[5] + D[6] + D[7];
    waveSum = __shfl(s, 0) + __shfl(s, 16);
#else
    float s = acc;
    for (int off = 16; off > 0; off >>= 1) s += __shfl_down(s, off);
    waveSum = __shfl(s, 0);
#endif

    __shared__ float wsum[8];
    int lane = threadIdx.x & 31, wv = threadIdx.x >> 5;
    if (lane == 0) wsum[wv] = waveSum;
    __syncthreads();
    if (threadIdx.x == 0) {
        float tot = 0.0f;
        for (int k = 0; k < 8; ++k) tot += wsum[k];
        blockSums[p] = tot;
    }
}

// ---------------- final combine ----------------
__global__ void k_final(const float* __restrict__ accum,
                        const float* __restrict__ blockSums,
                        float* __restrict__ out) {
    float tot = 0.0f;
    for (int k = 0; k < BDIM * NBANCH; ++k) tot += blockSums[k];
    const float n = (float)NOBJ;
    float localization = accum[0] / n;
    float conf_obj     = accum[1] / n;
    float class_loss   = accum[2] / n;
    float conf_nothing = tot / (float)NMASK;
    out[0] = 7.0f * localization + 5.0f * conf_obj + 5.0f * conf_nothing + class_loss;
}

extern "C" void kernel_launch(void* const* d_in, const int* in_sizes, int n_in,
                              void* d_out, int out_size, void* d_ws, size_t ws_size,
                              hipStream_t stream) {
    const float* P     = (const float*)d_in[0];   // predictions (8,190,96,96)
    const float* T     = (const float*)d_in[1];   // targets     (8,190,96,96)
    const int*   grids = (const int*)d_in[2];     // (256,4)

    char* ws = (char*)d_ws;
    float*         ious      = (float*)(ws + 0);        // 2560 f32
    int*           sel       = (int*)(ws + 10240);      // 256 i32
    int*           selxy     = (int*)(ws + 11264);      // 256 i32
    float*         accum     = (float*)(ws + 12288);    // loc, conf, cls
    float*         blockSums = (float*)(ws + 12352);    // 80 f32
    unsigned char* mask      = (unsigned char*)(ws + 12800); // 737280 bytes

    k_zero<<<(NMASK/4 + 255)/256, 256, 0, stream>>>((unsigned int*)mask, NMASK/4);
    k_iou<<<(NOBJ*NBANCH + 255)/256, 256, 0, stream>>>(P, T, grids, ious);
    k_scan<<<1, 32, 0, stream>>>(grids, ious, sel, selxy);
    k_gridloss<<<1, NOBJ, 0, stream>>>(P, T, grids, sel, selxy, mask, accum);
    k_bigred<<<BDIM*NBANCH, 256, 0, stream>>>(P, T, mask, blockSums);
    k_final<<<1, 1, 0, stream>>>(accum, blockSums, (float*)d_out);
}